// ConvLTVFilterGenerator_72868415144383
// MI455X (gfx1250) — compile-verified
//
#include <hip/hip_runtime.h>

// ---------------------------------------------------------------------------
// MI455X (gfx1250): convs as fp32 WMMA (v_wmma_f32_16x16x4_f32) with LDS-
// staged im2col tiles; cepstrum/correlation chain as LDS Stockham FFTs.
// ---------------------------------------------------------------------------

typedef __attribute__((ext_vector_type(2))) float v2f;
typedef __attribute__((ext_vector_type(8))) float v8f;

#define Bsz   8
#define Tsz   1000
#define INC   80
#define CONVC 256
#define CCEP  222
#define FFTN  1024
#define HOP   256
#define WIN   512
#define PADC  401          // (1024-222)/2
#define NFFT  2048
#define ZLEN  (Tsz*HOP)    // 256000

// ===========================================================================
// conv1: (8000 x 240) x (240 x 256) + ReLU.
// K-order: k = q*80 + ci (tap-major).  Block = 8 waves; A tile staged in LDS.
// WMMA f32 16x16x4 fragment layout:
//   A: lanes0-15 -> {K0,K1}, lanes16-31 -> {K2,K3} (2 VGPRs)
//   B: mirrored; C/D: vgpr v -> M = v + 8*(lane>=16), N = lane&15
// ===========================================================================
__global__ __launch_bounds__(256) void conv1_kernel(
    const float* __restrict__ X, const float* __restrict__ W,
    const float* __restrict__ bias, float* __restrict__ H) {
  __shared__ float Alds[16][241];          // 240 + 1 pad (odd dword stride)
  const int tid  = threadIdx.x;
  const int wave = tid >> 5;
  const int lane = tid & 31;
  const int half = lane >> 4;
  const int l16  = lane & 15;
  const int m0   = blockIdx.x * 16;

  // ---- stage im2col tile (boundary zeros baked in) ----
  for (int i = tid; i < 16 * 240; i += 256) {
    int row = i / 240, col = i - row * 240;
    int q = col / 80, ci = col - q * 80;
    int m = m0 + row;
    int b = m / Tsz, t = m - b * Tsz;
    int tt = t + q - 1;
    float v = 0.f;
    if (tt >= 0 && tt < Tsz) v = X[(b * Tsz + tt) * INC + ci];
    Alds[row][col] = v;
  }
  __syncthreads();

  // ---- each wave: one 16x16 N tile ----
  const int ntile = blockIdx.y * 8 + wave;     // 0..15
  const int n = ntile * 16 + l16;
  v8f acc = {0.f, 0.f, 0.f, 0.f, 0.f, 0.f, 0.f, 0.f};
  for (int q = 0; q < 3; ++q) {
    const float* wq = W + n * 240 + q;         // W[n][ci][q], stride 3 in ci
    #pragma unroll 4
    for (int ci = 0; ci < 80; ci += 4) {
      int k = q * 80 + ci + half * 2;
      v2f a, bb;
      a.x  = Alds[l16][k];
      a.y  = Alds[l16][k + 1];
      bb.x = wq[(ci + half * 2) * 3];
      bb.y = wq[(ci + half * 2) * 3 + 3];
      acc = __builtin_amdgcn_wmma_f32_16x16x4_f32(false, a, false, bb,
                                                  (short)0, acc, false, false);
    }
  }
  const float bv = bias[n];
  #pragma unroll
  for (int v = 0; v < 8; ++v) {
    int row = m0 + v + half * 8;
    float val = acc[v] + bv;
    H[row * CONVC + n] = val > 0.f ? val : 0.f;
  }
}

// ===========================================================================
// conv2/conv3 (8 groups): per group (8000 x 96) x (96 x 32) + ReLU.
// Block covers 8 N tiles = 4 groups = 128 input channels; A staged in LDS
// with K-order k = q*128 + cl (cl = local channel within the 128-chunk).
// ===========================================================================
__global__ __launch_bounds__(256) void convg_kernel(
    const float* __restrict__ Hin, const float* __restrict__ W,
    const float* __restrict__ bias, float* __restrict__ Hout) {
  __shared__ float Alds[16][385];          // 384 + 1 pad
  const int tid  = threadIdx.x;
  const int wave = tid >> 5;
  const int lane = tid & 31;
  const int half = lane >> 4;
  const int l16  = lane & 15;
  const int m0   = blockIdx.x * 16;
  const int cbase = blockIdx.y * 128;      // first input channel staged

  for (int i = tid; i < 16 * 384; i += 256) {
    int row = i / 384, col = i - row * 384;
    int q = col >> 7, cl = col & 127;
    int m = m0 + row;
    int b = m / Tsz, t = m - b * Tsz;
    int tt = t + q - 1;
    float v = 0.f;
    if (tt >= 0 && tt < Tsz) v = Hin[(b * Tsz + tt) * CONVC + cbase + cl];
    Alds[row][col] = v;
  }
  __syncthreads();

  const int ntile = blockIdx.y * 8 + wave; // 0..15
  const int gl = wave >> 1;                // group within staged chunk (0..3)
  const int n = ntile * 16 + l16;          // output channel
  v8f acc = {0.f, 0.f, 0.f, 0.f, 0.f, 0.f, 0.f, 0.f};
  for (int q = 0; q < 3; ++q) {
    const float* wq = W + n * 96 + q;      // W[n][il][q], stride 3 in il
    #pragma unroll
    for (int il = 0; il < 32; il += 4) {
      int k = q * 128 + gl * 32 + il + half * 2;
      v2f a, bb;
      a.x  = Alds[l16][k];
      a.y  = Alds[l16][k + 1];
      bb.x = wq[(il + half * 2) * 3];
      bb.y = wq[(il + half * 2) * 3 + 3];
      acc = __builtin_amdgcn_wmma_f32_16x16x4_f32(false, a, false, bb,
                                                  (short)0, acc, false, false);
    }
  }
  const float bv = bias[n];
  #pragma unroll
  for (int v = 0; v < 8; ++v) {
    int row = m0 + v + half * 8;
    float val = acc[v] + bv;
    Hout[row * CONVC + n] = val > 0.f ? val : 0.f;
  }
}

// ===========================================================================
// conv4: (8000 x 768) x (768 x 222); epilogue adds bias and divides by
// quefrency weights.  K-order k = q*256 + ci.  14 N tiles over grid.y = 2
// (last 2 waves of the second block slice idle through compute).
// ===========================================================================
__global__ __launch_bounds__(256) void conv4_kernel(
    const float* __restrict__ Hin, const float* __restrict__ W,
    const float* __restrict__ bias, float* __restrict__ C) {
  __shared__ float Alds[16][769];          // 768 + 1 pad (48.1 KB)
  const int tid  = threadIdx.x;
  const int wave = tid >> 5;
  const int lane = tid & 31;
  const int half = lane >> 4;
  const int l16  = lane & 15;
  const int m0   = blockIdx.x * 16;

  for (int i = tid; i < 16 * 768; i += 256) {
    int row = i / 768, col = i - row * 768;
    int q = col >> 8, ci = col & 255;
    int m = m0 + row;
    int b = m / Tsz, t = m - b * Tsz;
    int tt = t + q - 1;
    float v = 0.f;
    if (tt >= 0 && tt < Tsz) v = Hin[(b * Tsz + tt) * CONVC + ci];
    Alds[row][col] = v;
  }
  __syncthreads();

  const int ntile = blockIdx.y * 8 + wave;     // 0..15, need 0..13
  if (ntile < 14) {                            // wave-uniform branch
    const int n = ntile * 16 + l16;            // up to 223
    const bool nvalid = (n < CCEP);
    const int nsafe = nvalid ? n : 0;
    v8f acc = {0.f, 0.f, 0.f, 0.f, 0.f, 0.f, 0.f, 0.f};
    for (int q = 0; q < 3; ++q) {
      const float* wq = W + nsafe * 768 + q;   // W[n][ci][q], stride 3
      #pragma unroll 4
      for (int ci = 0; ci < 256; ci += 4) {
        int k = q * 256 + ci + half * 2;
        v2f a, bb;
        a.x  = Alds[l16][k];
        a.y  = Alds[l16][k + 1];
        bb.x = nvalid ? wq[(ci + half * 2) * 3]     : 0.f;
        bb.y = nvalid ? wq[(ci + half * 2) * 3 + 3] : 0.f;
        acc = __builtin_amdgcn_wmma_f32_16x16x4_f32(false, a, false, bb,
                                                    (short)0, acc, false,
                                                    false);
      }
    }
    if (nvalid) {
      const float bv = bias[n];
      const float quef = (n < 111) ? (float)(111 - n) : (float)(n - 110);
      const float inv_q = 1.f / quef;
      #pragma unroll
      for (int v = 0; v < 8; ++v) {
        int row = m0 + v + half * 8;
        C[row * CCEP + n] = (acc[v] + bv) * inv_q;
      }
    }
  }
}

// ===========================================================================
// Radix-2 Stockham complex FFT in LDS (autosort, ping-pong), 256 threads.
// sign = -1 forward (numpy convention), +1 inverse (caller scales by 1/N).
// Result guaranteed back in xr/xi.
// ===========================================================================
__device__ void block_fft(float* xr, float* xi, float* tr, float* ti,
                          int N, float sign) {
  const int tid = threadIdx.x;
  const int nth = 256;
  float *ar = xr, *ai = xi, *br = tr, *bi = ti;
  __syncthreads();
  const int halfN = N >> 1;
  int p = 0;
  for (int Ns = 1; Ns < N; Ns <<= 1, ++p) {
    for (int j = tid; j < halfN; j += nth) {
      int jm = j & (Ns - 1);
      float ang = sign * 3.14159265358979323846f * (float)jm / (float)Ns;
      float s, c;
      __sincosf(ang, &s, &c);
      float ur = ar[j],         ui = ai[j];
      float pr = ar[j + halfN], pi = ai[j + halfN];
      float vr = pr * c - pi * s;
      float vi = pr * s + pi * c;
      int idxD = ((j >> p) << (p + 1)) + jm;
      br[idxD]      = ur + vr;  bi[idxD]      = ui + vi;
      br[idxD + Ns] = ur - vr;  bi[idxD + Ns] = ui - vi;
    }
    __syncthreads();
    float* tp;
    tp = ar; ar = br; br = tp;
    tp = ai; ai = bi; bi = tp;
  }
  if (ar != xr) {   // odd #stages (N=2048): copy back
    for (int j = tid; j < N; j += nth) { xr[j] = ar[j]; xi[j] = ai[j]; }
    __syncthreads();
  }
}

// ===========================================================================
// Per-(b,t) pipeline: cepstrum FFT -> 10^Re * e^{i Im} -> impulse -> FFT
// cross-correlation with frame of z -> Hann window -> l to out, r to rbuf.
// One workgroup per (b,t); 52 KB LDS.
// ===========================================================================
__global__ __launch_bounds__(256) void fft_pipeline_kernel(
    const float* __restrict__ ccep,   // (8000, 222), already / quef
    const float* __restrict__ z,      // (8, 256000)
    float* __restrict__ out,          // (8, 256000)  <- l part
    float* __restrict__ rbuf) {       // (8000, 256)  <- r part
  __shared__ float xr[NFFT], xi[NFFT], tr[NFFT], ti[NFFT];
  __shared__ float fyr[NFFT], fyi[NFFT];
  __shared__ float imp[FFTN];

  const int tid = threadIdx.x;
  const int bt = blockIdx.x;
  const int b = bt / Tsz, t = bt - b * Tsz;

  // ---- Phase A: Y = fft1024(pad(ccep)); spec = 10^Re * e^{i Im}; imp = ifft
  for (int j = tid; j < FFTN; j += 256) {
    int k = j - PADC;
    xr[j] = (k >= 0 && k < CCEP) ? ccep[bt * CCEP + k] : 0.f;
    xi[j] = 0.f;
  }
  block_fft(xr, xi, tr, ti, FFTN, -1.f);
  for (int j = tid; j < FFTN; j += 256) {
    float mag = __expf(xr[j] * 2.302585092994046f);   // 10^x
    float s, c;
    __sincosf(xi[j], &s, &c);
    xr[j] = mag * c;
    xi[j] = mag * s;
  }
  block_fft(xr, xi, tr, ti, FFTN, +1.f);
  for (int j = tid; j < FFTN; j += 256)
    imp[j] = xr[j] * (1.f / (float)FFTN);
  __syncthreads();

  // ---- Phase B: Fy = fft2048(imp zero-padded), cached in fyr/fyi
  for (int j = tid; j < NFFT; j += 256) {
    xr[j] = (j < FFTN) ? imp[j] : 0.f;
    xi[j] = 0.f;
  }
  block_fft(xr, xi, tr, ti, NFFT, -1.f);
  for (int j = tid; j < NFFT; j += 256) { fyr[j] = xr[j]; fyi[j] = xi[j]; }

  // ---- Phase C: Fz = fft2048(frame placed at offset 1023)
  for (int j = tid; j < NFFT; j += 256) {
    float v = 0.f;
    int i = j - (NFFT / 2 - 1);          // 1023
    if (i >= 0 && i < WIN) {
      int zi = t * HOP + i - (WIN / 2 - 1);   // z padded by (255, 256)
      if (zi >= 0 && zi < ZLEN) v = z[b * ZLEN + zi];
    }
    xr[j] = v;
    xi[j] = 0.f;
  }
  block_fft(xr, xi, tr, ti, NFFT, -1.f);

  // ---- Phase D: corr = ifft2048(Fz * conj(Fy)).real[:512]; window; scatter
  __syncthreads();
  for (int j = tid; j < NFFT; j += 256) {
    float zr = xr[j], zim = xi[j];
    float yr = fyr[j], yi2 = fyi[j];
    float pr = zr * yr + zim * yi2;
    float pi = zim * yr - zr * yi2;
    xr[j] = pr;
    xi[j] = pi;
  }
  block_fft(xr, xi, tr, ti, NFFT, +1.f);
  for (int m = tid; m < WIN; m += 256) {
    float corr = xr[m] * (1.f / (float)NFFT);
    float w = 0.5f * (1.f - __cosf(6.283185307179586f * (float)m /
                                   (float)WIN));
    float v = corr * w;
    if (m < HOP) out[b * ZLEN + t * HOP + m] = v;
    else         rbuf[bt * HOP + (m - HOP)] = v;
  }
}

// ===========================================================================
// Overlap-add: out[b, t*HOP + i] += r[b, (t-1) mod T, i]
// ===========================================================================
__global__ __launch_bounds__(256) void combine_kernel(
    float* __restrict__ out, const float* __restrict__ rbuf) {
  int idx = blockIdx.x * 256 + threadIdx.x;
  if (idx >= Bsz * ZLEN) return;
  int b = idx / ZLEN;
  int rem = idx - b * ZLEN;
  int t = rem / HOP;
  int i = rem - t * HOP;
  int tp = (t + Tsz - 1) % Tsz;
  out[idx] += rbuf[(b * Tsz + tp) * HOP + i];
}

// ---------------------------------------------------------------------------
extern "C" void kernel_launch(void* const* d_in, const int* in_sizes, int n_in,
                              void* d_out, int out_size, void* d_ws,
                              size_t ws_size, hipStream_t stream) {
  const float* x  = (const float*)d_in[0];
  const float* z  = (const float*)d_in[1];
  const float* W1 = (const float*)d_in[2];
  const float* b1 = (const float*)d_in[3];
  const float* W2 = (const float*)d_in[4];
  const float* b2 = (const float*)d_in[5];
  const float* W3 = (const float*)d_in[6];
  const float* b3 = (const float*)d_in[7];
  const float* W4 = (const float*)d_in[8];
  const float* b4 = (const float*)d_in[9];
  float* out = (float*)d_out;

  float* ws = (float*)d_ws;
  float* h1 = ws;                              // 8000*256
  float* h2 = h1 + (size_t)Bsz * Tsz * CONVC;  // 8000*256
  float* cc = h2 + (size_t)Bsz * Tsz * CONVC;  // 8000*222
  float* rb = cc + (size_t)Bsz * Tsz * CCEP;   // 8000*256

  const int MT = (Bsz * Tsz) / 16;   // 500 M-tiles
  conv1_kernel<<<dim3(MT, 2), 256, 0, stream>>>(x, W1, b1, h1);
  convg_kernel<<<dim3(MT, 2), 256, 0, stream>>>(h1, W2, b2, h2);
  convg_kernel<<<dim3(MT, 2), 256, 0, stream>>>(h2, W3, b3, h1);
  conv4_kernel<<<dim3(MT, 2), 256, 0, stream>>>(h1, W4, b4, cc);
  fft_pipeline_kernel<<<Bsz * Tsz, 256, 0, stream>>>(cc, z, out, rb);
  combine_kernel<<<(Bsz * ZLEN + 255) / 256, 256, 0, stream>>>(out, rb);
}